// BiLSTMWindowModel_11699490914385
// MI455X (gfx1250) — compile-verified
//
#include <hip/hip_runtime.h>
#include <hip/hip_bf16.h>
#include <math.h>

// Problem constants
#define Bsz 512
#define Tn  200
#define Din 40
#define Hn  256

typedef __attribute__((ext_vector_type(2))) float v2f;
typedef __attribute__((ext_vector_type(8))) float v8f;

// D = A(16x4 f32) * B(4x16 f32) + C(16x16 f32), full fp32 WMMA (CDNA5)
__device__ __forceinline__ v8f wmma_k4(v2f a, v2f b, v8f c) {
  return __builtin_amdgcn_wmma_f32_16x16x4_f32(
      /*neg_a=*/false, a, /*neg_b=*/false, b,
      /*c_mod=*/(short)0, c, /*reuse_a=*/false, /*reuse_b=*/false);
}

__device__ __forceinline__ float sigm(float v) { return 1.0f / (1.0f + expf(-v)); }

// ---------------------------------------------------------------------------
// Init: xv[b,t,0:40] = tanh(x), xv[b,t,40:80] = sigmoid(x); zero state block
// ---------------------------------------------------------------------------
__global__ void init_kernel(const float* __restrict__ x, float* __restrict__ xv,
                            float* __restrict__ states, int nstates) {
  int i = blockIdx.x * blockDim.x + threadIdx.x;
  if (i < Bsz * Tn * Din) {
    int bt = i / Din;
    int k  = i - bt * Din;
    float v = x[i];
    xv[(size_t)bt * (2 * Din) + k]       = tanhf(v);
    xv[(size_t)bt * (2 * Din) + Din + k] = sigm(v);
  }
  if (i < nstates) states[i] = 0.0f;
}

// ---------------------------------------------------------------------------
// Layer-0 LSTM step (both directions via blockIdx.z).
// Block = 128 threads = 4 waves; wave g computes gate-g 16x32 tile
// (two 16x16 WMMA column tiles, A-fragment reused -> 3 loads / 2 WMMA).
// grid = (B/16=32 row tiles, H/32=8 col tiles, 2 dirs)
// ---------------------------------------------------------------------------
__global__ void lstm_step_l0(
    const float* __restrict__ xv,                       // [B, T, 80]
    const float* __restrict__ w_ih_f, const float* __restrict__ w_hh_f,
    const float* __restrict__ b_ih_f, const float* __restrict__ b_hh_f,
    const float* __restrict__ w_ih_r, const float* __restrict__ w_hh_r,
    const float* __restrict__ b_ih_r, const float* __restrict__ b_hh_r,
    const float* __restrict__ h_in,                     // [2][B*H]
    float* __restrict__ h_out,                          // [2][B*H]
    float* __restrict__ c_state,                        // [2][B*H]
    float* __restrict__ h0,                             // [B, T, 512]
    int step) {
  const int dir = blockIdx.z;
  const int t   = dir ? (Tn - 1 - step) : step;
  const float* w_ih = dir ? w_ih_r : w_ih_f;
  const float* w_hh = dir ? w_hh_r : w_hh_f;
  const float* b_ih = dir ? b_ih_r : b_ih_f;
  const float* b_hh = dir ? b_hh_r : b_hh_f;
  const float* hprev = h_in + (size_t)dir * (Bsz * Hn);

  const int wave  = threadIdx.x >> 5;      // gate id 0..3 (i,f,g,o)
  const int lane  = threadIdx.x & 31;
  const int ln15  = lane & 15;
  const int khalf = (lane >> 4) * 2;       // K sub-offset {0,2}
  const int rowt  = blockIdx.x;            // batch tile (16 rows)
  const int colt  = blockIdx.y;            // hidden-col tile (32 cols)

  const int arow  = rowt * 16 + ln15;                 // batch row (A frag)
  const int jcol0 = wave * Hn + colt * 32 + ln15;     // gate column, tile 0
  const int jcol1 = jcol0 + 16;                       // gate column, tile 1

  const float* xrow = xv + (size_t)arow * (Tn * 2 * Din) + (size_t)t * (2 * Din);
  const float* hrow = hprev + (size_t)arow * Hn;
  const float* wi0  = w_ih + (size_t)jcol0 * (2 * Din);
  const float* wi1  = w_ih + (size_t)jcol1 * (2 * Din);
  const float* wh0  = w_hh + (size_t)jcol0 * Hn;
  const float* wh1  = w_hh + (size_t)jcol1 * Hn;

  v8f acc0 = {}, acc1 = {};
  // Input contribution: K = 0..79
  #pragma unroll 4
  for (int k0 = 0; k0 < 2 * Din; k0 += 4) {
    v2f a, b0, b1;
    a[0]  = xrow[k0 + khalf];  a[1]  = xrow[k0 + khalf + 1];
    b0[0] = wi0[k0 + khalf];   b0[1] = wi0[k0 + khalf + 1];
    b1[0] = wi1[k0 + khalf];   b1[1] = wi1[k0 + khalf + 1];
    acc0 = wmma_k4(a, b0, acc0);
    acc1 = wmma_k4(a, b1, acc1);
  }
  // Recurrent contribution: K = 0..255
  #pragma unroll 4
  for (int k0 = 0; k0 < Hn; k0 += 4) {
    v2f a, b0, b1;
    a[0]  = hrow[k0 + khalf];  a[1]  = hrow[k0 + khalf + 1];
    b0[0] = wh0[k0 + khalf];   b0[1] = wh0[k0 + khalf + 1];
    b1[0] = wh1[k0 + khalf];   b1[1] = wh1[k0 + khalf + 1];
    acc0 = wmma_k4(a, b0, acc0);
    acc1 = wmma_k4(a, b1, acc1);
  }

  // Dump gate tiles to LDS per C/D layout: VGPR r -> rows {r, r+8}
  __shared__ float g_lds[4][16][32];
  const int nloc  = lane & 15;
  const int mbase = (lane >> 4) * 8;
  #pragma unroll
  for (int r = 0; r < 8; ++r) {
    g_lds[wave][mbase + r][nloc]      = acc0[r];
    g_lds[wave][mbase + r][nloc + 16] = acc1[r];
  }
  __syncthreads();

  // Combine gates: 16x32 = 512 elements / 128 threads
  for (int e = threadIdx.x; e < 512; e += 128) {
    const int em = e >> 5, en = e & 31;
    const int b  = rowt * 16 + em;
    const int ng = colt * 32 + en;
    float gi = g_lds[0][em][en] + b_ih[ng]            + b_hh[ng];
    float gf = g_lds[1][em][en] + b_ih[Hn + ng]       + b_hh[Hn + ng];
    float gg = g_lds[2][em][en] + b_ih[2 * Hn + ng]   + b_hh[2 * Hn + ng];
    float go = g_lds[3][em][en] + b_ih[3 * Hn + ng]   + b_hh[3 * Hn + ng];
    float si = sigm(gi), sf = sigm(gf), tg = tanhf(gg), so = sigm(go);
    float* cp = c_state + (size_t)dir * (Bsz * Hn) + (size_t)b * Hn + ng;
    float cn = sf * (*cp) + si * tg;
    *cp = cn;
    float hn = so * tanhf(cn);
    h_out[(size_t)dir * (Bsz * Hn) + (size_t)b * Hn + ng] = hn;
    h0[(size_t)b * (Tn * 2 * Hn) + (size_t)t * (2 * Hn) + dir * Hn + ng] = hn;
  }
}

// ---------------------------------------------------------------------------
// Layer-1 LSTM step (single direction; weights passed in). din = 512.
// grid = (32, 8), block = 128
// ---------------------------------------------------------------------------
__global__ void lstm_step_l1(
    const float* __restrict__ h0, int t,                // [B, T, 512]
    const float* __restrict__ w_ih, const float* __restrict__ w_hh,
    const float* __restrict__ b_ih, const float* __restrict__ b_hh,
    const float* __restrict__ h_in,                     // [B*H]
    float* __restrict__ h_out,                          // [B*H]
    float* __restrict__ c_state) {                      // [B*H]
  const int wave  = threadIdx.x >> 5;
  const int lane  = threadIdx.x & 31;
  const int ln15  = lane & 15;
  const int khalf = (lane >> 4) * 2;
  const int rowt  = blockIdx.x;
  const int colt  = blockIdx.y;

  const int arow  = rowt * 16 + ln15;
  const int jcol0 = wave * Hn + colt * 32 + ln15;
  const int jcol1 = jcol0 + 16;

  const float* xrow = h0 + (size_t)arow * (Tn * 2 * Hn) + (size_t)t * (2 * Hn);
  const float* hrow = h_in + (size_t)arow * Hn;
  const float* wi0  = w_ih + (size_t)jcol0 * (2 * Hn);
  const float* wi1  = w_ih + (size_t)jcol1 * (2 * Hn);
  const float* wh0  = w_hh + (size_t)jcol0 * Hn;
  const float* wh1  = w_hh + (size_t)jcol1 * Hn;

  v8f acc0 = {}, acc1 = {};
  #pragma unroll 4
  for (int k0 = 0; k0 < 2 * Hn; k0 += 4) {    // K = 512 input part
    v2f a, b0, b1;
    a[0]  = xrow[k0 + khalf];  a[1]  = xrow[k0 + khalf + 1];
    b0[0] = wi0[k0 + khalf];   b0[1] = wi0[k0 + khalf + 1];
    b1[0] = wi1[k0 + khalf];   b1[1] = wi1[k0 + khalf + 1];
    acc0 = wmma_k4(a, b0, acc0);
    acc1 = wmma_k4(a, b1, acc1);
  }
  #pragma unroll 4
  for (int k0 = 0; k0 < Hn; k0 += 4) {        // K = 256 recurrent part
    v2f a, b0, b1;
    a[0]  = hrow[k0 + khalf];  a[1]  = hrow[k0 + khalf + 1];
    b0[0] = wh0[k0 + khalf];   b0[1] = wh0[k0 + khalf + 1];
    b1[0] = wh1[k0 + khalf];   b1[1] = wh1[k0 + khalf + 1];
    acc0 = wmma_k4(a, b0, acc0);
    acc1 = wmma_k4(a, b1, acc1);
  }

  __shared__ float g_lds[4][16][32];
  const int nloc  = lane & 15;
  const int mbase = (lane >> 4) * 8;
  #pragma unroll
  for (int r = 0; r < 8; ++r) {
    g_lds[wave][mbase + r][nloc]      = acc0[r];
    g_lds[wave][mbase + r][nloc + 16] = acc1[r];
  }
  __syncthreads();

  for (int e = threadIdx.x; e < 512; e += 128) {
    const int em = e >> 5, en = e & 31;
    const int b  = rowt * 16 + em;
    const int ng = colt * 32 + en;
    float gi = g_lds[0][em][en] + b_ih[ng]          + b_hh[ng];
    float gf = g_lds[1][em][en] + b_ih[Hn + ng]     + b_hh[Hn + ng];
    float gg = g_lds[2][em][en] + b_ih[2 * Hn + ng] + b_hh[2 * Hn + ng];
    float go = g_lds[3][em][en] + b_ih[3 * Hn + ng] + b_hh[3 * Hn + ng];
    float si = sigm(gi), sf = sigm(gf), tg = tanhf(gg), so = sigm(go);
    float* cp = c_state + (size_t)b * Hn + ng;
    float cn = sf * (*cp) + si * tg;
    *cp = cn;
    h_out[(size_t)b * Hn + ng] = so * tanhf(cn);
  }
}

// ---------------------------------------------------------------------------
// Head: out[b,d] = head_b[d] + sum_k last[b,k] * head_w[d,k]
// last = [h1f_final | h1r_first]
// ---------------------------------------------------------------------------
__global__ void head_kernel(const float* __restrict__ h1f, const float* __restrict__ h1r,
                            const float* __restrict__ head_w, const float* __restrict__ head_b,
                            float* __restrict__ out) {
  int idx = blockIdx.x * blockDim.x + threadIdx.x;   // b*2 + d
  if (idx >= Bsz * 2) return;
  int b = idx >> 1, d = idx & 1;
  float s = head_b[d];
  const float* wf = head_w + (size_t)d * (2 * Hn);
  for (int k = 0; k < Hn; ++k) s += h1f[(size_t)b * Hn + k] * wf[k];
  for (int k = 0; k < Hn; ++k) s += h1r[(size_t)b * Hn + k] * wf[Hn + k];
  out[idx] = s;
}

// ---------------------------------------------------------------------------
extern "C" void kernel_launch(void* const* d_in, const int* in_sizes, int n_in,
                              void* d_out, int out_size, void* d_ws, size_t ws_size,
                              hipStream_t stream) {
  const float* x        = (const float*)d_in[0];
  const float* w_ih_l0  = (const float*)d_in[1];
  const float* w_hh_l0  = (const float*)d_in[2];
  const float* b_ih_l0  = (const float*)d_in[3];
  const float* b_hh_l0  = (const float*)d_in[4];
  const float* w_ih_l0r = (const float*)d_in[5];
  const float* w_hh_l0r = (const float*)d_in[6];
  const float* b_ih_l0r = (const float*)d_in[7];
  const float* b_hh_l0r = (const float*)d_in[8];
  const float* w_ih_l1  = (const float*)d_in[9];
  const float* w_hh_l1  = (const float*)d_in[10];
  const float* b_ih_l1  = (const float*)d_in[11];
  const float* b_hh_l1  = (const float*)d_in[12];
  const float* w_ih_l1r = (const float*)d_in[13];
  const float* w_hh_l1r = (const float*)d_in[14];
  const float* b_ih_l1r = (const float*)d_in[15];
  const float* b_hh_l1r = (const float*)d_in[16];
  const float* head_w   = (const float*)d_in[17];
  const float* head_b   = (const float*)d_in[18];

  // Workspace layout (floats)
  float* ws = (float*)d_ws;
  const size_t N_XV = (size_t)Bsz * Tn * (2 * Din);          // 8,192,000
  const size_t SBH  = (size_t)Bsz * Hn;                      // 131,072
  float* xv      = ws;
  float* states  = ws + N_XV;
  float* l0h[2]  = { states,            states + 2 * SBH };  // [2][2 dirs][B*H]
  float* l0c     = states + 4 * SBH;                         // [2 dirs][B*H]
  float* l1h[2]  = { states + 6 * SBH,  states + 7 * SBH };
  float* l1c     = states + 8 * SBH;
  float* l1rzero = states + 9 * SBH;                         // stays zero
  float* l1rc    = states + 10 * SBH;
  float* l1rh    = states + 11 * SBH;
  const int NSTATES = (int)(12 * SBH);
  float* h0      = states + 12 * SBH;                        // [B, T, 512]

  // 1) xv + zero state buffers
  {
    int n = Bsz * Tn * Din;                                  // 4,096,000
    init_kernel<<<(n + 255) / 256, 256, 0, stream>>>(x, xv, states, NSTATES);
  }

  // 2) Layer 0, both directions, 200 steps
  for (int s = 0; s < Tn; ++s) {
    lstm_step_l0<<<dim3(Bsz / 16, Hn / 32, 2), 128, 0, stream>>>(
        xv, w_ih_l0, w_hh_l0, b_ih_l0, b_hh_l0,
        w_ih_l0r, w_hh_l0r, b_ih_l0r, b_hh_l0r,
        l0h[s & 1], l0h[(s + 1) & 1], l0c, h0, s);
  }

  // 3) Layer 1 forward, 200 steps (only final h needed)
  for (int s = 0; s < Tn; ++s) {
    lstm_step_l1<<<dim3(Bsz / 16, Hn / 32), 128, 0, stream>>>(
        h0, s, w_ih_l1, w_hh_l1, b_ih_l1, b_hh_l1,
        l1h[s & 1], l1h[(s + 1) & 1], l1c);
  }
  // Layer 1 backward: hs[T-1] is the state after processing only x[T-1] -> 1 step
  lstm_step_l1<<<dim3(Bsz / 16, Hn / 32), 128, 0, stream>>>(
      h0, Tn - 1, w_ih_l1r, w_hh_l1r, b_ih_l1r, b_hh_l1r,
      l1rzero, l1rh, l1rc);

  // 4) Head. Layer-1 fwd final state: after step s=199, out buffer = l1h[0].
  head_kernel<<<(Bsz * 2 + 255) / 256, 256, 0, stream>>>(
      l1h[0], l1rh, head_w, head_b, (float*)d_out);
}